// Gating_65592740545200
// MI455X (gfx1250) — compile-verified
//
#include <hip/hip_runtime.h>
#include <hip/hip_bf16.h>
#include <math.h>

typedef __attribute__((ext_vector_type(2))) float v2f;
typedef __attribute__((ext_vector_type(8))) float v8f;

#define D_FEAT 128
#define NHID_  256
#define KREAL  385   // 3*128 + 1
#define KPAD   388   // padded to multiple of 4 for WMMA K-steps
#define KPAIR  (KPAD / 2)

// ---------------------------------------------------------------------------
// Zero the accumulator workspace (AX, AX2, deg live contiguously in d_ws).
// ---------------------------------------------------------------------------
__global__ __launch_bounds__(256) void zero_ws_kernel(float* __restrict__ p, long n) {
  long i = (long)blockIdx.x * blockDim.x + threadIdx.x;
  long stride = (long)gridDim.x * blockDim.x;
  for (; i < n; i += stride) p[i] = 0.0f;
}

// ---------------------------------------------------------------------------
// Build pair-packed, zero-padded w1pk: for even row r,
//   w1pk[(r/2)*256 + col] = { w1[r][col], w1[r+1][col] }
// so each WMMA B-fragment is a single coalesced global_load_b64.
// ---------------------------------------------------------------------------
__global__ __launch_bounds__(256) void pack_w1_kernel(const float* __restrict__ w1,
                                                      float2* __restrict__ w1pk) {
  int i = blockIdx.x * blockDim.x + threadIdx.x;
  int stride = gridDim.x * blockDim.x;
  for (; i < KPAIR * NHID_; i += stride) {
    int pr  = i >> 8;        // pair index
    int col = i & 255;
    int r   = pr << 1;
    float2 v;
    v.x = (r     < KREAL) ? w1[r       * NHID_ + col] : 0.0f;
    v.y = (r + 1 < KREAL) ? w1[(r + 1) * NHID_ + col] : 0.0f;
    w1pk[i] = v;
  }
}

// ---------------------------------------------------------------------------
// Phase 1: one wave per edge. 32 lanes x float4 = one coalesced 512B row of x.
// Accumulate val*x and val*x^2 with HW fp32 atomics (L2-resident targets).
// ---------------------------------------------------------------------------
__global__ __launch_bounds__(256) void edge_scatter_kernel(
    const float* __restrict__ x, const int* __restrict__ er,
    const int* __restrict__ ec, const float* __restrict__ ev,
    float* __restrict__ AX, float* __restrict__ AX2, float* __restrict__ deg,
    int E) {
  int gid   = blockIdx.x * blockDim.x + threadIdx.x;
  int wave  = gid >> 5;
  int lane  = threadIdx.x & 31;
  int nwav  = (gridDim.x * blockDim.x) >> 5;
  for (int e = wave; e < E; e += nwav) {
    int   r = er[e];
    int   c = ec[e];
    float v = ev[e];
    const float4* xp = (const float4*)(x + (size_t)c * D_FEAT);
    float4 xv = xp[lane];                       // global_load_b128, fully coalesced
    float* axp  = AX  + (size_t)r * D_FEAT + lane * 4;
    float* ax2p = AX2 + (size_t)r * D_FEAT + lane * 4;
    unsafeAtomicAdd(axp  + 0, v * xv.x);
    unsafeAtomicAdd(axp  + 1, v * xv.y);
    unsafeAtomicAdd(axp  + 2, v * xv.z);
    unsafeAtomicAdd(axp  + 3, v * xv.w);
    unsafeAtomicAdd(ax2p + 0, v * xv.x * xv.x);
    unsafeAtomicAdd(ax2p + 1, v * xv.y * xv.y);
    unsafeAtomicAdd(ax2p + 2, v * xv.z * xv.z);
    unsafeAtomicAdd(ax2p + 3, v * xv.w * xv.w);
    if (lane == 0) unsafeAtomicAdd(deg + r, v);
  }
}

// ---------------------------------------------------------------------------
// Phase 2: per 16-node slab. Build x_cat in LDS, WMMA f32 16x16x4 GEMM against
// pair-packed w1pk (L2-resident), LayerNorm + ReLU, 256x2 projection, softmax.
// Block = 256 threads = 8 waves; each wave owns two 16-column tiles (8*2*16=256).
// ---------------------------------------------------------------------------
__global__ __launch_bounds__(256) void node_mlp_kernel(
    const float* __restrict__ x,   const float* __restrict__ AX,
    const float* __restrict__ AX2, const float* __restrict__ deg,
    const float2* __restrict__ w1pk, const float* __restrict__ b1,
    const float* __restrict__ ln_g, const float* __restrict__ ln_b,
    const float* __restrict__ w2,  const float* __restrict__ b2,
    float* __restrict__ out, int N) {
  __shared__ float sA[16 * KPAD];    // x_cat slab, K padded with zeros
  __shared__ float sH[16 * NHID_];   // pre-LN hidden activations
  __shared__ float sPart[2 * 256];   // LN / logit partials
  __shared__ float sMu[16];
  __shared__ float sRs[16];

  const int node0 = blockIdx.x * 16;
  const int tid   = threadIdx.x;

  // ---- Stage x_cat = [x, AX - x, x_std, deg, 0-pad] into LDS ----
  for (int idx = tid; idx < 16 * KPAD; idx += 256) {
    int m = idx / KPAD;
    int j = idx - m * KPAD;
    int node = node0 + m;
    float val = 0.0f;
    if (node < N) {
      if (j < 128) {
        val = x[(size_t)node * D_FEAT + j];
      } else if (j < 256) {
        int f = j - 128;
        val = AX[(size_t)node * D_FEAT + f] - x[(size_t)node * D_FEAT + f];
      } else if (j < 384) {
        int f = j - 256;
        float dg   = deg[node];
        float inv  = 1.0f / (dg + 1e-8f);
        float mean = AX[(size_t)node * D_FEAT + f] * inv;
        float var  = AX2[(size_t)node * D_FEAT + f] * inv - mean * mean;
        val = sqrtf(fmaxf(var, 0.0f));
      } else if (j == 384) {
        val = deg[node];
      } // j in [385,388) stays 0 (K padding)
    }
    sA[idx] = val;
  }
  __syncthreads();

  // ---- WMMA f32 16x16x4: h = x_cat @ w1 + b1  (branch-free K-loop) ----
  const int w    = tid >> 5;          // wave id 0..7
  const int lane = tid & 31;
  const int mn   = lane & 15;         // M row (A) / N col (B,D) for this lane
  const int kh   = (lane >> 4) << 1;  // K sub-offset: lanes 0-15 -> 0, 16-31 -> 2
  const v2f* w1v = (const v2f*)w1pk;
  for (int t = 0; t < 2; ++t) {
    const int colbase = (w + t * 8) * 16;
    const int col     = colbase + mn;
    v8f c = {0.f, 0.f, 0.f, 0.f, 0.f, 0.f, 0.f, 0.f};
    #pragma unroll 4
    for (int k = 0; k < KPAD; k += 4) {
      const int r0 = k + kh;                              // always even
      v2f a = *(const v2f*)&sA[mn * KPAD + r0];           // ds_load_b64 (8B aligned)
      v2f b = w1v[(r0 >> 1) * NHID_ + col];               // one global_load_b64
      c = __builtin_amdgcn_wmma_f32_16x16x4_f32(
          /*neg_a=*/false, a, /*neg_b=*/false, b,
          /*c_mod=*/(short)0, c, /*reuse_a=*/false, /*reuse_b=*/false);
    }
    // D layout: VGPR i -> M = i (lanes 0-15) / M = 8+i (lanes 16-31), N = col
    const int mbase = (lane >> 4) * 8;
    const float bb  = b1[col];
    #pragma unroll
    for (int i = 0; i < 8; ++i)
      sH[(mbase + i) * NHID_ + col] = c[i] + bb;
  }
  __syncthreads();

  // ---- LayerNorm statistics: 16 threads per row, 16-element partials ----
  {
    int m = tid >> 4;           // row 0..15
    int i = tid & 15;           // chunk 0..15
    float s = 0.0f, s2 = 0.0f;
    #pragma unroll
    for (int j = 0; j < 16; ++j) {
      float h = sH[m * NHID_ + i * 16 + j];
      s += h; s2 += h * h;
    }
    sPart[tid]       = s;
    sPart[256 + tid] = s2;
  }
  __syncthreads();
  if (tid < 16) {
    float s = 0.0f, s2 = 0.0f;
    #pragma unroll
    for (int i = 0; i < 16; ++i) {
      s  += sPart[tid * 16 + i];
      s2 += sPart[256 + tid * 16 + i];
    }
    float mu  = s * (1.0f / NHID_);
    float var = s2 * (1.0f / NHID_) - mu * mu;
    sMu[tid] = mu;
    sRs[tid] = rsqrtf(fmaxf(var, 0.0f) + 1e-5f);
  }
  __syncthreads();

  // ---- LN + ReLU + 256x2 projection (16 threads per row) + softmax ----
  {
    int m = tid >> 4;           // row 0..15
    int i = tid & 15;           // chunk 0..15
    float mu = sMu[m], rs = sRs[m];
    float l0 = 0.0f, l1 = 0.0f;
    #pragma unroll
    for (int j = 0; j < 16; ++j) {
      int jj = i * 16 + j;
      float h = (sH[m * NHID_ + jj] - mu) * rs * ln_g[jj] + ln_b[jj];
      h = fmaxf(h, 0.0f);
      l0 += h * w2[jj * 2 + 0];
      l1 += h * w2[jj * 2 + 1];
    }
    sPart[tid]       = l0;
    sPart[256 + tid] = l1;
  }
  __syncthreads();
  if (tid < 16) {
    int node = node0 + tid;
    if (node < N) {
      float l0 = b2[0], l1 = b2[1];
      #pragma unroll
      for (int i = 0; i < 16; ++i) {
        l0 += sPart[tid * 16 + i];
        l1 += sPart[256 + tid * 16 + i];
      }
      float mx = fmaxf(l0, l1);
      float e0 = __expf(l0 - mx), e1 = __expf(l1 - mx);
      float inv = 1.0f / (e0 + e1);
      out[(size_t)node * 2 + 0] = e0 * inv;
      out[(size_t)node * 2 + 1] = e1 * inv;
    }
  }
}

// ---------------------------------------------------------------------------
extern "C" void kernel_launch(void* const* d_in, const int* in_sizes, int n_in,
                              void* d_out, int out_size, void* d_ws, size_t ws_size,
                              hipStream_t stream) {
  const float* x    = (const float*)d_in[0];
  const int*   er   = (const int*)  d_in[1];
  const int*   ec   = (const int*)  d_in[2];
  const float* ev   = (const float*)d_in[3];
  const float* w1   = (const float*)d_in[4];
  const float* b1   = (const float*)d_in[5];
  const float* ln_g = (const float*)d_in[6];
  const float* ln_b = (const float*)d_in[7];
  const float* w2   = (const float*)d_in[8];
  const float* b2   = (const float*)d_in[9];
  float* out = (float*)d_out;

  const int N = in_sizes[0] / D_FEAT;   // 50000
  const int E = in_sizes[1];            // 1600000

  // Workspace layout: AX [N*128] | AX2 [N*128] | deg [N] | w1pk [KPAIR*256 float2]
  float*  AX   = (float*)d_ws;
  float*  AX2  = AX  + (size_t)N * D_FEAT;
  float*  deg  = AX2 + (size_t)N * D_FEAT;
  float2* w1pk = (float2*)(deg + (size_t)N);

  long zn = (long)N * (2 * D_FEAT) + N;
  zero_ws_kernel<<<1024, 256, 0, stream>>>(AX, zn);
  pack_w1_kernel<<<128, 256, 0, stream>>>(w1, w1pk);
  edge_scatter_kernel<<<4096, 256, 0, stream>>>(x, er, ec, ev, AX, AX2, deg, E);
  node_mlp_kernel<<<(N + 15) / 16, 256, 0, stream>>>(
      x, AX, AX2, deg, w1pk, b1, ln_g, ln_b, w2, b2, out, N);
}